// VectorQuantize_31714038514117
// MI455X (gfx1250) — compile-verified
//
#include <hip/hip_runtime.h>

#define DIM   512
#define NEMB  8192
#define NTOK  32768
#define KB    16          // k-block staged per LDS buffer
#define LDA   20          // padded A row stride in floats (bank- and align-safe)

typedef float v2f __attribute__((ext_vector_type(2)));
typedef float v8f __attribute__((ext_vector_type(8)));

// Monotone order-preserving float->uint encoding (for packed u64 atomic argmin).
__device__ __forceinline__ unsigned int fenc(float f) {
    unsigned int u = __float_as_uint(f);
    return (u & 0x80000000u) ? ~u : (u | 0x80000000u);
}

// One 16-byte async global->LDS transfer per lane (CDNA5 VGLOBAL async op).
__device__ __forceinline__ void async_copy16(unsigned lds_off, const float* gptr) {
    asm volatile("global_load_async_to_lds_b128 %0, %1, off"
                 :: "v"(lds_off), "v"(gptr)
                 : "memory");
}

__device__ __forceinline__ void wait_async0() {
    asm volatile("s_wait_asynccnt 0x0" ::: "memory");
}

// ---------------- init: packed (dist,col) argmin slots to +inf ----------------
__global__ void vq_init(unsigned long long* __restrict__ amin) {
    int i = blockIdx.x * 256 + threadIdx.x;
    amin[i] = 0xFFFFFFFFFFFFFFFFull;
}

// ---------------- per-token squared L2 norms: one wave per row ----------------
__global__ void vq_row_norms(const float* __restrict__ x, float* __restrict__ xn) {
    int row  = blockIdx.x * 8 + (threadIdx.x >> 5);
    int lane = threadIdx.x & 31;
    const float* p = x + (size_t)row * DIM;
    float s = 0.f;
    for (int k = lane; k < DIM; k += 32) { float v = p[k]; s += v * v; }
    #pragma unroll
    for (int m = 16; m >= 1; m >>= 1) s += __shfl_xor(s, m, 32);
    if (lane == 0) xn[row] = s;
}

// ---------------- per-codeword squared L2 norms (column sums) -----------------
__global__ void vq_col_norms(const float* __restrict__ e, float* __restrict__ en) {
    int n = blockIdx.x * 256 + threadIdx.x;
    float s = 0.f;
    for (int k = 0; k < DIM; ++k) { float v = e[(size_t)k * NEMB + n]; s += v * v; }
    en[n] = s;
}

// ------- fused dist GEMM: async-LDS double buffering + fp32 WMMA + argmin -----
// Workgroup: 8 waves (2x4) -> 64 rows x 256 cols.  Wave: 32 rows x 64 cols
// = 2x4 accumulators of v_wmma_f32_16x16x4_f32, K staged through LDS in
// blocks of 16 via GLOBAL_LOAD_ASYNC_TO_LDS_B128 (double buffered, ASYNCcnt).
__global__ __launch_bounds__(256)
void vq_dist_gemm(const float* __restrict__ x, const float* __restrict__ e,
                  const float* __restrict__ xn, const float* __restrict__ en,
                  float* __restrict__ dist, unsigned long long* __restrict__ amin) {
    __shared__ float ldsA[2][64 * LDA];   // 64 rows x KB (padded)   ~10 KB
    __shared__ float ldsB[2][KB * 256];   // KB k-rows x 256 cols     32 KB

    const int tid  = threadIdx.x;
    const int lane = tid & 31;
    const int wave = tid >> 5;
    const int wm   = wave >> 2;        // 0..1 : 32-row group
    const int wn   = wave & 3;         // 0..3 : 64-col group
    const int half = lane >> 4;        // lane half
    const int l    = lane & 15;
    const int row0 = blockIdx.y * 64;
    const int col0 = blockIdx.x * 256;

    // Cooperative copy mapping (per thread): A = 1x16B chunk, B = 4x16B chunks.
    const int arow   = tid >> 2;           // 0..63
    const int achunk = (tid & 3) << 2;     // 0,4,8,12 floats

    auto issue = [&](int kb, int buf) {
        async_copy16((unsigned)(size_t)&ldsA[buf][arow * LDA + achunk],
                     x + (size_t)(row0 + arow) * DIM + kb * KB + achunk);
        #pragma unroll
        for (int c = 0; c < 4; ++c) {
            int krow = c * 4 + (tid >> 6);     // 0..15
            int colc = (tid & 63) << 2;        // 0..252 floats
            async_copy16((unsigned)(size_t)&ldsB[buf][krow * 256 + colc],
                         e + (size_t)(kb * KB + krow) * NEMB + col0 + colc);
        }
    };

    v8f acc[2][4];
    v8f z = {};
    #pragma unroll
    for (int mt = 0; mt < 2; ++mt)
        #pragma unroll
        for (int t = 0; t < 4; ++t) acc[mt][t] = z;

    issue(0, 0);

    for (int kb = 0; kb < DIM / KB; ++kb) {
        const int buf = kb & 1;
        wait_async0();           // own async writes for buf done
        __syncthreads();         // everyone's writes done; prev buffer fully read
        if (kb + 1 < DIM / KB) issue(kb + 1, buf ^ 1);

        #pragma unroll
        for (int kk = 0; kk < KB; kk += 4) {
            v2f a[2];
            #pragma unroll
            for (int mt = 0; mt < 2; ++mt)
                a[mt] = *(const v2f*)&ldsA[buf][(wm * 32 + mt * 16 + l) * LDA + kk + half * 2];
            v2f b[4];
            #pragma unroll
            for (int t = 0; t < 4; ++t) {
                b[t].x = ldsB[buf][(kk + half * 2    ) * 256 + wn * 64 + t * 16 + l];
                b[t].y = ldsB[buf][(kk + half * 2 + 1) * 256 + wn * 64 + t * 16 + l];
            }
            #pragma unroll
            for (int mt = 0; mt < 2; ++mt)
                #pragma unroll
                for (int t = 0; t < 4; ++t)
                    acc[mt][t] = __builtin_amdgcn_wmma_f32_16x16x4_f32(
                        false, a[mt], false, b[t], (short)0, acc[mt][t], false, false);
        }
    }

    // Epilogue: dist = ||x||^2 + ||e||^2 - 2*dot ; store + fused argmin.
    float env[4];
    #pragma unroll
    for (int t = 0; t < 4; ++t) env[t] = en[col0 + wn * 64 + t * 16 + l];

    #pragma unroll
    for (int mt = 0; mt < 2; ++mt) {
        const int rbase = row0 + wm * 32 + mt * 16 + half * 8;
        #pragma unroll
        for (int j = 0; j < 8; ++j) {
            const float xnv = xn[rbase + j];
            float best = 0.f;
            unsigned int bcol = 0;
            #pragma unroll
            for (int t = 0; t < 4; ++t) {
                unsigned int cc = (unsigned int)(col0 + wn * 64 + t * 16 + l);
                float d = xnv + env[t] - 2.0f * acc[mt][t][j];
                dist[(size_t)(rbase + j) * NEMB + cc] = d;
                if (t == 0 || d < best) { best = d; bcol = cc; }  // ties keep lower col
            }
            unsigned long long p = ((unsigned long long)fenc(best) << 32) | bcol;
            #pragma unroll
            for (int m = 8; m >= 1; m >>= 1) {
                unsigned int hi = __shfl_xor((unsigned int)(p >> 32), m, 32);
                unsigned int lo = __shfl_xor((unsigned int)p, m, 32);
                unsigned long long q = ((unsigned long long)hi << 32) | lo;
                if (q < p) p = q;
            }
            if (l == 0) atomicMin(&amin[rbase + j], p);
        }
    }
}

// ------------- gather quantize, emit indices, per-row loss partials -----------
__global__ void vq_gather(const float* __restrict__ x, const float* __restrict__ e,
                          const unsigned long long* __restrict__ amin,
                          float* __restrict__ out_q, float* __restrict__ out_ind,
                          float* __restrict__ lpart) {
    const int row = blockIdx.x;
    const int t   = threadIdx.x;
    const int ind = (int)(amin[row] & 0xFFFFFFFFull);
    float s = 0.f;
    for (int k = t; k < DIM; k += 256) {
        float q = e[(size_t)k * NEMB + ind];
        out_q[(size_t)row * DIM + k] = q;
        float d = x[(size_t)row * DIM + k] - q;
        s += d * d;
    }
    __shared__ float red[256];
    red[t] = s;
    __syncthreads();
    for (int off = 128; off > 0; off >>= 1) {
        if (t < off) red[t] += red[t + off];
        __syncthreads();
    }
    if (t == 0) {
        lpart[row]   = red[0];
        out_ind[row] = (float)ind;
    }
}

// ---------------- deterministic fixed-order loss reduction --------------------
__global__ void vq_loss(const float* __restrict__ lpart, float* __restrict__ out_loss) {
    const int t = threadIdx.x;
    float s = 0.f;
    for (int i = t; i < NTOK; i += 256) s += lpart[i];
    __shared__ float red[256];
    red[t] = s;
    __syncthreads();
    for (int off = 128; off > 0; off >>= 1) {
        if (t < off) red[t] += red[t + off];
        __syncthreads();
    }
    if (t == 0) out_loss[0] = red[0] * (1.0f / ((float)NTOK * (float)DIM));
}

extern "C" void kernel_launch(void* const* d_in, const int* in_sizes, int n_in,
                              void* d_out, int out_size, void* d_ws, size_t ws_size,
                              hipStream_t stream) {
    const float* input = (const float*)d_in[0];   // [NTOK, DIM]
    const float* embed = (const float*)d_in[1];   // [DIM, NEMB]

    float* out      = (float*)d_out;
    float* out_q    = out;                              // NTOK*DIM
    float* out_ind  = out + (size_t)NTOK * DIM;         // NTOK
    float* out_loss = out_ind + NTOK;                   // 1
    float* out_dist = out_loss + 1;                     // NTOK*NEMB

    float* xn = (float*)d_ws;                           // NTOK f32
    float* en = xn + NTOK;                              // NEMB f32
    unsigned long long* amin = (unsigned long long*)(en + NEMB);  // NTOK u64 (8B aligned)
    float* lpart = (float*)(amin + NTOK);               // NTOK f32

    hipLaunchKernelGGL(vq_init,      dim3(NTOK / 256),            dim3(256), 0, stream, amin);
    hipLaunchKernelGGL(vq_row_norms, dim3(NTOK / 8),              dim3(256), 0, stream, input, xn);
    hipLaunchKernelGGL(vq_col_norms, dim3(NEMB / 256),            dim3(256), 0, stream, embed, en);
    hipLaunchKernelGGL(vq_dist_gemm, dim3(NEMB / 256, NTOK / 64), dim3(256), 0, stream,
                       input, embed, xn, en, out_dist, amin);
    hipLaunchKernelGGL(vq_gather,    dim3(NTOK),                  dim3(256), 0, stream,
                       input, embed, amin, out_q, out_ind, lpart);
    hipLaunchKernelGGL(vq_loss,      dim3(1),                     dim3(256), 0, stream,
                       lpart, out_loss);
}